// ApplyBasisCLIMB_38268158607916
// MI455X (gfx1250) — compile-verified
//
#include <hip/hip_runtime.h>

typedef float v2f __attribute__((ext_vector_type(2)));
typedef float v8f __attribute__((ext_vector_type(8)));

#define NTERMS 128
#define NPIX   768
#define PPSZ   256
#define PLANE  (NPIX * NPIX)          // 589824 elements per basis plane

// ---------------------------------------------------------------------------
// Kernel 1: L[i*768 + j] = sum_n coeffs[n] * basis[n*PLANE + i*768 + j]
//
// Bandwidth-bound (302 MB of basis @ 23.3 TB/s => ~13 us floor). Compute is
// free, so we drive it through V_WMMA_F32_16X16X4_F32 at full f32 precision:
//   A (16x4)  = coeffs chunk replicated across all 16 rows
//               (A layout: lane<16 holds M=lane,K={0,1}; lane>=16 K={2,3};
//                i.e. k = 2*half + vgpr  -> a = {c[n0+2h], c[n0+2h+1]})
//   B (4x16)  = basis rows n0..n0+3 for 16 consecutive pixels
//               (same k = 2*half + vgpr striping; lanes 0-15 carry N=pixel)
//   D (16x16) = every row m equals the 16 partial latent values -> take M=0.
// One wave owns one 16-pixel group; K=128 fully unrolled as 32 chained WMMAs
// with 64 independent b32 loads (every fetched byte is used).
// ---------------------------------------------------------------------------
__global__ __launch_bounds__(256) void
einsum_wmma_kernel(const float* __restrict__ basis,
                   const float* __restrict__ coeffs,
                   float* __restrict__ L)
{
    const int lane  = threadIdx.x & 31;
    const int wave  = (int)((blockIdx.x * blockDim.x + threadIdx.x) >> 5);
    const int group = wave;                       // one 16-pixel group / wave
    if (group >= PLANE / 16) return;              // wave-uniform guard

    const int  h    = lane >> 4;                  // lane half (0/1)
    const int  nlo  = lane & 15;                  // N index within group
    const long pix  = (long)group * 16 + nlo;     // this lane's pixel column

    const float* bcol = basis + pix;

    v8f acc = {};
#pragma unroll
    for (int t = 0; t < 32; ++t) {                // K chunk: n0 = 4*t
        const int n0 = 4 * t + 2 * h;

        v2f a;                                    // A[m][k] = c[4t + 2h + v]
        a.x = coeffs[n0 + 0];
        a.y = coeffs[n0 + 1];

        v2f b;                                    // B[k][n] = basis[4t+2h+v][pix]
        b.x = bcol[(long)(n0 + 0) * PLANE];
        b.y = bcol[(long)(n0 + 1) * PLANE];

        // D = A*B + C  (chained f32 accumulation, no precision loss)
        acc = __builtin_amdgcn_wmma_f32_16x16x4_f32(
                  /*neg_a=*/false, a, /*neg_b=*/false, b,
                  /*c_mod=*/(short)0, acc, /*reuse_a=*/false, /*reuse_b=*/false);
    }

    // D VGPR0, lanes 0-15: M=0, N=lane  ->  latent for pixel group*16+lane
    if (h == 0) {
        L[pix] = acc[0];
    }
}

// ---------------------------------------------------------------------------
// Kernel 2: CLIMB area per 3x3 patch. out[cy][ax] uses
//   flat[b*3+d] = wf[ax*3+b][cy*3+d] = L[(cy*3+d)*768 + (ax*3+b)]
// LSQ matrix rows derived analytically from the reference's meshgrid:
//   a = (f2+f5+f8-f0-f3-f6)/3,  b = (f6+f7+f8-f0-f1-f2)/3,
//   c = 4/9*f0 + 5/18*(f1+f3) + 1/9*(f2+f4+f6) - 1/18*(f5+f7) - 2/9*f8
// where-ladder applied in the reference's exact order; final scale is
// pi*d * wavel/(2*pi) = d * wavel * 0.5.
// ---------------------------------------------------------------------------
__global__ __launch_bounds__(256) void
climb_kernel(const float* __restrict__ L,
             const float* __restrict__ wavel,
             float* __restrict__ out)
{
    const int idx = blockIdx.x * blockDim.x + threadIdx.x;   // 0..65535
    if (idx >= PPSZ * PPSZ) return;
    const int ax = idx & (PPSZ - 1);     // output column ("a" index)
    const int cy = idx >> 8;             // output row    ("c" index)

    float f[9];
#pragma unroll
    for (int b = 0; b < 3; ++b)
#pragma unroll
        for (int d = 0; d < 3; ++d)
            f[b * 3 + d] = L[(cy * 3 + d) * NPIX + (ax * 3 + b)];

    const float EPS = 1e-15f;
    float pa = (f[2] + f[5] + f[8] - f[0] - f[3] - f[6]) * (1.0f / 3.0f);
    float pb = (f[6] + f[7] + f[8] - f[0] - f[1] - f[2]) * (1.0f / 3.0f);
    float pc = (4.0f / 9.0f) * f[0]
             + (5.0f / 18.0f) * (f[1] + f[3])
             + (1.0f / 9.0f) * (f[2] + f[4] + f[6])
             - (1.0f / 18.0f) * (f[5] + f[7])
             - (2.0f / 9.0f) * f[8];

    pa = (pa == 0.0f) ? EPS : pa;
    pb = (pb == 0.0f) ? EPS : pb;
    pc = (pc == 0.0f) ? EPS : pc;

    float x1 = (-pb - pc) / pa;
    float x2 = -pc / pa;
    float lo = fminf(x1, x2);
    float hi = fmaxf(x1, x2);
    x1 = fmaxf(lo, 0.0f);
    x2 = fminf(hi, 1.0f);

    float cb = -pc / pb;
    float ab = 0.5f * (pa / pb);
    float d  = x1 + cb * x2 - ab * x2 * x2 - cb * x1 + ab * x1 * x1;

    d = (d >= 0.5f) ? d : 1.0f - d;

    float s = f[0] + f[1] + f[2] + f[3] + f[4] + f[5] + f[6] + f[7] + f[8];
    d = (s >= 0.0f) ? d : 1.0f - d;     // mean(flat) >= 0  <=>  sum >= 0

    bool allpos = (f[0] > 0.0f) && (f[1] > 0.0f) && (f[2] > 0.0f) &&
                  (f[3] > 0.0f) && (f[4] > 0.0f) && (f[5] > 0.0f) &&
                  (f[6] > 0.0f) && (f[7] > 0.0f) && (f[8] > 0.0f);
    bool allnp  = (f[0] <= 0.0f) && (f[1] <= 0.0f) && (f[2] <= 0.0f) &&
                  (f[3] <= 0.0f) && (f[4] <= 0.0f) && (f[5] <= 0.0f) &&
                  (f[6] <= 0.0f) && (f[7] <= 0.0f) && (f[8] <= 0.0f);
    bool anyz   = (f[0] == 0.0f) || (f[1] == 0.0f) || (f[2] == 0.0f) ||
                  (f[3] == 0.0f) || (f[4] == 0.0f) || (f[5] == 0.0f) ||
                  (f[6] == 0.0f) || (f[7] == 0.0f) || (f[8] == 0.0f);

    d = allpos ? 1.0f : d;
    d = allnp  ? 0.0f : d;
    d = anyz   ? ((d > 0.0f) ? 1.0f : 0.0f) : d;

    d = fminf(fmaxf(d, 0.0f), 1.0f);

    out[idx] = d * wavel[0] * 0.5f;     // pi*d * wavel / (2*pi)
}

// ---------------------------------------------------------------------------
extern "C" void kernel_launch(void* const* d_in, const int* in_sizes, int n_in,
                              void* d_out, int out_size, void* d_ws, size_t ws_size,
                              hipStream_t stream)
{
    const float* basis  = (const float*)d_in[0];   // (128, 768, 768) f32
    const float* coeffs = (const float*)d_in[1];   // (128,) f32
    const float* wavel  = (const float*)d_in[2];   // scalar f32
    float*       out    = (float*)d_out;           // (256, 256) f32
    float*       L      = (float*)d_ws;            // 768*768 f32 scratch (2.36 MB)

    // Kernel 1: 36864 waves (one 16-pixel group each) = 4608 blocks x 256 thr.
    const int groups = PLANE / 16;                 // 36864
    const int blocks1 = groups / 8;                // 8 waves per 256-thr block
    einsum_wmma_kernel<<<blocks1, 256, 0, stream>>>(basis, coeffs, L);

    // Kernel 2: one thread per output pixel.
    climb_kernel<<<(PPSZ * PPSZ) / 256, 256, 0, stream>>>(L, wavel, out);
}